// Part_Graph_51539607552364
// MI455X (gfx1250) — compile-verified
//
#include <hip/hip_runtime.h>

typedef __attribute__((ext_vector_type(2))) float v2f;
typedef __attribute__((ext_vector_type(8))) float v8f;

#define HD   10
#define PN   6
#define HWPX 16384   // 128*128
#define HIN  64

// LDS layout (floats)
#define WUP_OFF  0      // [6][4 blocks][16 rows][12 k] = 4608
#define WDP_OFF  4608   // [5][2 blocks][16][12]        = 1920
#define WFA_OFF  6528   // [6][16]
#define WFB_OFF  6624
#define WHA_OFF  6720
#define WHB_OFF  6816
#define WATT_OFF 6912
#define BIAS_OFF 7008   // bf[6], bh[6], batt[6]
#define XPT_OFF  7040   // per-wave xp tiles: [8 waves][6 parts][12 rows][16 px]
#define XPT_WAVE 1152
#define LDS_N    (XPT_OFF + 8 * XPT_WAVE)   // 16256 floats = 65024 bytes

__device__ __forceinline__ float shx16(float x) { return __shfl_xor(x, 16, 32); }

__device__ __forceinline__ float fast_sigmoid(float x) {
  // v_exp_f32 + v_rcp_f32: avoids the IEEE div_scale/fma fixup chain
  return __builtin_amdgcn_rcpf(1.f + __expf(-x));
}

__device__ __forceinline__ v8f wmma4(v2f a, v2f b, v8f c) {
  // D(16x16 f32) = A(16x4 f32) x B(4x16 f32) + C   — exact f32 accumulate
  return __builtin_amdgcn_wmma_f32_16x16x4_f32(false, a, false, b, (short)0, c,
                                               false, false);
}

// A fragment: weights. lane holds row m = lane&15, K = k0 + 2*half + {0,1}
__device__ __forceinline__ v2f load_a(const float* sw, int aoff, int row, int kk) {
  v2f a;
  a[0] = sw[aoff + row * 12 + kk];
  a[1] = sw[aoff + row * 12 + kk + 1];
  return a;
}

// B fragment straight from an LDS activation tile [12 rows][16 px]:
//   B VGPR j = row (k0 + j + 2*half), N = pixel (lane&15). Rows 10,11 are zero.
__device__ __forceinline__ v2f load_b_lds(const float* sw, int tbase, int cb,
                                          int half, int l16) {
  const int base = tbase + (cb * 4 + 2 * half) * 16 + l16;
  v2f b;
  b[0] = sw[base];
  b[1] = sw[base + 16];
  return b;
}

// B fragment from a register C-layout vector (lane holds pixel lane&15,
// channels v + 8*half; >=10 are zero) via cross-half shuffles.
__device__ __forceinline__ v2f make_b(const float* v, int cb, int half) {
  float b0, b1;
  if (cb == 0) {
    float t0 = shx16(v[2]), t1 = shx16(v[3]);
    b0 = half ? t0 : v[0];  b1 = half ? t1 : v[1];
  } else if (cb == 1) {
    float t0 = shx16(v[6]), t1 = shx16(v[7]);
    b0 = half ? t0 : v[4];  b1 = half ? t1 : v[5];
  } else {
    float t0 = shx16(v[0]), t1 = shx16(v[1]);      // ch 8,9 live on upper half
    b0 = half ? 0.f : t0;   b1 = half ? 0.f : t1;  // ch 10,11 are padding
  }
  v2f r; r[0] = b0; r[1] = b1; return r;
}

__global__ __launch_bounds__(256) void part_graph_fused(
    const float* __restrict__ g_xf,  const float* __restrict__ g_xh0,
    const float* __restrict__ g_xh1, const float* __restrict__ g_xp,
    const float* __restrict__ g_Wf,  const float* __restrict__ g_bf,
    const float* __restrict__ g_Wh,  const float* __restrict__ g_bh,
    const float* __restrict__ g_Wa,  const float* __restrict__ g_ba,
    const float* __restrict__ g_Wdp, const float* __restrict__ g_Wup,
    float* __restrict__ o_xp, float* __restrict__ o_att,
    int n_tiles, int n_waves)
{
  __shared__ float sw[LDS_N];
  // ---- cooperative weight stage into LDS (zero-padded 16x12 WMMA tiles) ----
  for (int i = threadIdx.x; i < XPT_OFF; i += blockDim.x) {
    float val = 0.f;
    if (i < WDP_OFF) {
      int k = i % 12, r = (i / 12) & 15, pb = i / 192, b = pb & 3, p = pb >> 2;
      if (r < HD && k < HD) val = g_Wup[(p * HD + r) * 40 + b * HD + k];
    } else if (i < WFA_OFF) {
      int j = i - WDP_OFF;
      int k = j % 12, r = (j / 12) & 15, eb = j / 192, b = eb & 1, e = eb >> 1;
      if (r < HD && k < HD) val = g_Wdp[(e * HD + r) * 20 + b * HD + k];
    } else if (i < WFB_OFF) { int j=i-WFA_OFF, c=j&15, p=j>>4; if (c<HD) val = g_Wf[p*20 + c]; }
    else if (i < WHA_OFF)   { int j=i-WFB_OFF, c=j&15, p=j>>4; if (c<HD) val = g_Wf[p*20 + HD + c]; }
    else if (i < WHB_OFF)   { int j=i-WHA_OFF, c=j&15, p=j>>4; if (c<HD) val = g_Wh[p*20 + c]; }
    else if (i < WATT_OFF)  { int j=i-WHB_OFF, c=j&15, p=j>>4; if (c<HD) val = g_Wh[p*20 + HD + c]; }
    else if (i < BIAS_OFF)  { int j=i-WATT_OFF,c=j&15, p=j>>4; if (c<HD) val = g_Wa[p*HD + c]; }
    else { int j=i-BIAS_OFF; if (j<6) val=g_bf[j]; else if (j<12) val=g_bh[j-6];
           else if (j<18) val=g_ba[j-12]; }
    sw[i] = val;
  }
  __syncthreads();

  const int lane = threadIdx.x & 31;
  const int half = lane >> 4;
  const int l16  = lane & 15;
  const int waveInBlk = threadIdx.x >> 5;
  const int wbase = XPT_OFF + waveInBlk * XPT_WAVE;
  const int waveId = (blockIdx.x * blockDim.x + threadIdx.x) >> 5;

  // zero-pad rows 10,11 of every part tile once (wave-private region)
#pragma unroll
  for (int p = 0; p < PN; ++p) sw[wbase + p * 192 + 160 + lane] = 0.f;

  // part -> incident edge lists: EDGES = (0,1),(1,2),(2,3),(1,4),(4,5)
  const int NINC[PN]      = {1, 3, 2, 1, 2, 1};
  const int INC_E[PN][3]  = {{0,0,0},{0,1,3},{1,2,0},{2,0,0},{3,4,0},{4,0,0}};
  const int INC_OT[PN][3] = {{1,0,0},{0,2,4},{1,3,0},{2,0,0},{1,5,0},{4,0,0}};
  const int INC_RL[PN][3] = {{0,0,0},{1,0,0},{1,0,0},{1,0,0},{1,0,0},{1,0,0}};

  for (int t = waveId; t < n_tiles; t += n_waves) {   // wave-uniform loop
    const int n   = t >> 10;                // 1024 16-pixel tiles per image
    const int hw0 = (t & 1023) << 4;
    const int h   = hw0 >> 7;
    const int w   = (hw0 & 127) + l16;
    const int pix = hw0 + l16;

    // ---- stage this wave's xp tile into LDS: [part][12 rows][16 px] ----
    const float* xpn = g_xp + (size_t)n * PN * HD * HWPX;
#pragma unroll
    for (int p = 0; p < PN; ++p)
#pragma unroll
      for (int rep = 0; rep < 5; ++rep) {
        int idx = rep * 32 + lane;          // 0..159 = 10ch x 16px, coalesced
        sw[wbase + p * 192 + idx] =
            xpn[(p * HD + (idx >> 4)) * HWPX + hw0 + (idx & 15)];
      }

    // nearest upsample of xf (64 -> 128), C-layout registers
    float xfv[8];
    const float* xfn = g_xf + (size_t)n * HD * HIN * HIN;
    const int fi = (h >> 1) * HIN + (w >> 1);
#pragma unroll
    for (int v = 0; v < 8; ++v) {
      int ch = v + 8 * half, chc = ch < HD ? ch : (HD - 1);
      float x = xfn[chc * HIN * HIN + fi];
      xfv[v] = (ch < HD) ? x : 0.f;
    }

    // bilinear (align_corners=True) upsample of xh0/xh1
    float posh = (float)h * (63.0f / 127.0f);
    int   ih   = (int)posh; if (ih > 62) ih = 62;
    float fh   = posh - (float)ih;
    float posw = (float)w * (63.0f / 127.0f);
    int   iw   = (int)posw; if (iw > 62) iw = 62;
    float fw   = posw - (float)iw;
    const int b00 = ih * HIN + iw;
    float xh0v[8], xh1v[8];
    const float* h0n = g_xh0 + (size_t)n * HD * HIN * HIN;
    const float* h1n = g_xh1 + (size_t)n * HD * HIN * HIN;
#pragma unroll
    for (int v = 0; v < 8; ++v) {
      int ch = v + 8 * half, chc = ch < HD ? ch : (HD - 1);
      int base = chc * HIN * HIN + b00;
      float a00 = h0n[base],       a01 = h0n[base + 1];
      float a10 = h0n[base + HIN], a11 = h0n[base + HIN + 1];
      float r0 = a00 + (a01 - a00) * fw, r1 = a10 + (a11 - a10) * fw;
      float x0 = r0 + (r1 - r0) * fh;
      a00 = h1n[base];       a01 = h1n[base + 1];
      a10 = h1n[base + HIN]; a11 = h1n[base + HIN + 1];
      r0 = a00 + (a01 - a00) * fw; r1 = a10 + (a11 - a10) * fw;
      float x1 = r0 + (r1 - r0) * fh;
      xh0v[v] = (ch < HD) ? x0 : 0.f;
      xh1v[v] = (ch < HD) ? x1 : 0.f;
    }

    // ---- attention scalars: per-lane partial dot + xor-16 butterfly ----
    float attF[PN], attH[PN], dpA[PN];
#pragma unroll
    for (int p = 0; p < PN; ++p) {
      float s1 = 0.f, s2 = 0.f, s3 = 0.f;
      const int tp = wbase + p * 192;
#pragma unroll
      for (int v = 0; v < 8; ++v) {
        int c16 = v + 8 * half;
        int chc = c16 < HD ? c16 : 0;              // weight is 0 for c16>=10
        float xpc  = sw[tp + chc * 16 + l16];
        float xh_c = (p < 4) ? xh0v[v] : xh1v[v];
        s1 += sw[WFA_OFF  + p * 16 + c16] * xfv[v]
            + sw[WFB_OFF  + p * 16 + c16] * xpc;
        s2 += sw[WHA_OFF  + p * 16 + c16] * xh_c
            + sw[WHB_OFF  + p * 16 + c16] * xpc;
        s3 += sw[WATT_OFF + p * 16 + c16] * xpc;
      }
      s1 += shx16(s1); s2 += shx16(s2); s3 += shx16(s3);  // both halves get sum
      attF[p] = fast_sigmoid(s1 + sw[BIAS_OFF + p]);
      attH[p] = fast_sigmoid(s2 + sw[BIAS_OFF + 6 + p]);
      dpA[p]  = fast_sigmoid(s3 + sw[BIAS_OFF + 12 + p]);
    }

    // ---- per part: recompute incident edge messages, then update conv ----
#pragma unroll
    for (int p = 0; p < PN; ++p) {
      float xpp[8];
#pragma unroll
      for (int v = 0; v < 8; ++v) xpp[v] = 0.f;

#pragma unroll
      for (int j = 0; j < 3; ++j) {
        if (j < NINC[p]) {
          const int e = INC_E[p][j], other = INC_OT[p][j], role = INC_RL[p][j];
          const int ta = wbase + (role ? other : p) * 192;  // first endpoint
          const int tb = wbase + (role ? p : other) * 192;  // second endpoint
          v8f acc = (v8f)0.f;
#pragma unroll
          for (int blk = 0; blk < 2; ++blk) {
            const int tsrc = blk ? tb : ta;
            const int aoff = WDP_OFF + (e * 2 + blk) * 192;
#pragma unroll
            for (int cb = 0; cb < 3; ++cb) {
              v2f A = load_a(sw, aoff, l16, cb * 4 + 2 * half);
              v2f B = load_b_lds(sw, tsrc, cb, half, l16);
              acc = wmma4(A, B, acc);
            }
          }
          const float g = 2.f - dpA[other];
#pragma unroll
          for (int v = 0; v < 8; ++v) {
            float dp = acc[v] > 0.f ? acc[v] : 0.f;
            xpp[v] += dp * g;
          }
        }
      }

      float dF[8], dH[8];
#pragma unroll
      for (int v = 0; v < 8; ++v) {
        float xh_c = (p < 4) ? xh0v[v] : xh1v[v];
        dF[v] = attF[p] * xfv[v];
        dH[v] = attH[p] * xh_c;
      }

      v8f acc = (v8f)0.f;
      const int tp  = wbase + p * 192;
      const int aob = WUP_OFF + p * 4 * 192;
#pragma unroll
      for (int cb = 0; cb < 3; ++cb) {              // block 0: xp from LDS
        v2f A = load_a(sw, aob, l16, cb * 4 + 2 * half);
        v2f B = load_b_lds(sw, tp, cb, half, l16);
        acc = wmma4(A, B, acc);
      }
#pragma unroll
      for (int blk = 1; blk < 4; ++blk) {           // blocks 1-3: dF, dH, xpp
        const float* src = (blk == 1) ? dF : (blk == 2) ? dH : xpp;
        const int aoff = aob + blk * 192;
#pragma unroll
        for (int cb = 0; cb < 3; ++cb) {
          v2f A = load_a(sw, aoff, l16, cb * 4 + 2 * half);
          v2f B = make_b(src, cb, half);
          acc = wmma4(A, B, acc);
        }
      }

      // epilogue: xp_new = relu(xp + relu(upd)), coalesced C-layout store
      const size_t obase = (size_t)(n * PN + p) * HD * HWPX;
#pragma unroll
      for (int v = 0; v < 8; ++v) {
        int ch = v + 8 * half;
        int chc = ch < HD ? ch : 0;
        float xpold = sw[tp + chc * 16 + l16];
        float upd = acc[v] > 0.f ? acc[v] : 0.f;
        float xnew = xpold + upd;
        xnew = xnew > 0.f ? xnew : 0.f;
        if (ch < HD) o_xp[obase + (size_t)ch * HWPX + pix] = xnew;
      }
      if (!half) {
        float ap = (attF[p] + attH[p] + dpA[p]) * (1.f / 3.f);
        o_att[(size_t)(n * PN + p) * HWPX + pix] = ap;
      }
    }
  }
}

extern "C" void kernel_launch(void* const* d_in, const int* in_sizes, int n_in,
                              void* d_out, int out_size, void* d_ws, size_t ws_size,
                              hipStream_t stream) {
  const float* xf  = (const float*)d_in[0];
  const float* xh0 = (const float*)d_in[1];
  const float* xh1 = (const float*)d_in[2];
  const float* xp  = (const float*)d_in[3];
  const float* Wf  = (const float*)d_in[4];
  const float* bf  = (const float*)d_in[5];
  const float* Wh  = (const float*)d_in[6];
  const float* bh  = (const float*)d_in[7];
  const float* Wa  = (const float*)d_in[8];
  const float* ba  = (const float*)d_in[9];
  const float* Wdp = (const float*)d_in[10];
  const float* Wup = (const float*)d_in[11];

  const int N = in_sizes[3] / (PN * HD * HWPX);   // 32
  float* o_xp  = (float*)d_out;
  float* o_att = o_xp + (size_t)N * PN * HD * HWPX;

  const int n_tiles = N * (HWPX / 16);            // one 16-pixel tile per wave-iter
  int blocks = 1024;
  if (blocks * 8 > n_tiles) blocks = (n_tiles + 7) / 8;
  const int threads = 256;                        // 8 wave32 per block
  const int n_waves = blocks * (threads / 32);

  part_graph_fused<<<blocks, threads, 0, stream>>>(
      xf, xh0, xh1, xp, Wf, bf, Wh, bh, Wa, ba, Wdp, Wup,
      o_xp, o_att, n_tiles, n_waves);
}